// LinearAttention_67800353734986
// MI455X (gfx1250) — compile-verified
//
#include <hip/hip_runtime.h>
#include <hip/hip_bf16.h>
#include <math.h>

#define B_ 4
#define L_ 4096
#define D_ 1024
#define H_ 16
#define HD_ 64
#define M_TOTAL (B_ * L_) /* 16384 */

#define TM 128   /* GEMM block tile M */
#define TNB 128  /* GEMM block tile N (wide to cut L2 re-read traffic) */
#define TK 32
#define KVCHUNK 512

typedef float v2f __attribute__((ext_vector_type(2)));
typedef float v8f __attribute__((ext_vector_type(8)));

__device__ __forceinline__ v8f wmma_f32(v2f a, v2f b, v8f c) {
  // D = A(16x4 f32) x B(4x16 f32) + C(16x16 f32)   -> v_wmma_f32_16x16x4_f32
  return __builtin_amdgcn_wmma_f32_16x16x4_f32(false, a, false, b, (short)0, c,
                                               false, false);
}

// ---------------------------------------------------------------------------
// 0) zero the atomic accumulators (kv + ksum) every launch
// ---------------------------------------------------------------------------
__global__ void zero_kernel(float* __restrict__ p, int n) {
  int i = blockIdx.x * blockDim.x + threadIdx.x;
  if (i < n) p[i] = 0.0f;
}

// ---------------------------------------------------------------------------
// 1) LayerNorm: one block per row of 1024, 256 threads x 4 elems
// ---------------------------------------------------------------------------
__global__ __launch_bounds__(256) void ln_kernel(const float* __restrict__ x,
                                                 const float* __restrict__ gamma,
                                                 const float* __restrict__ beta,
                                                 float* __restrict__ xn) {
  __shared__ float red[256];
  const int row = blockIdx.x;
  const int tid = threadIdx.x;
  const float4 v = *reinterpret_cast<const float4*>(x + (size_t)row * D_ + tid * 4);
  red[tid] = v.x + v.y + v.z + v.w;
  __syncthreads();
  for (int off = 128; off > 0; off >>= 1) {
    if (tid < off) red[tid] += red[tid + off];
    __syncthreads();
  }
  const float mean = red[0] * (1.0f / D_);
  __syncthreads();
  const float d0 = v.x - mean, d1 = v.y - mean, d2 = v.z - mean, d3 = v.w - mean;
  red[tid] = d0 * d0 + d1 * d1 + d2 * d2 + d3 * d3;
  __syncthreads();
  for (int off = 128; off > 0; off >>= 1) {
    if (tid < off) red[tid] += red[tid + off];
    __syncthreads();
  }
  const float rstd = rsqrtf(red[0] * (1.0f / D_) + 1e-5f);
  const float4 g = *reinterpret_cast<const float4*>(gamma + tid * 4);
  const float4 b = *reinterpret_cast<const float4*>(beta + tid * 4);
  float4 o;
  o.x = d0 * rstd * g.x + b.x;
  o.y = d1 * rstd * g.y + b.y;
  o.z = d2 * rstd * g.z + b.z;
  o.w = d3 * rstd * g.w + b.w;
  *reinterpret_cast<float4*>(xn + (size_t)row * D_ + tid * 4) = o;
}

// ---------------------------------------------------------------------------
// 2) QKV GEMM: [16384,1024] @ [1024,3072], fused phi() on q/k,
//    scatter into q/k/v buffers laid out [B,H,L,64].
//    Block tile 128x128, 8 waves in 4(M)x2(N), each wave 32x64 (2x4 WMMA).
// ---------------------------------------------------------------------------
__global__ __launch_bounds__(256) void qkv_gemm_kernel(
    const float* __restrict__ A, const float* __restrict__ W,
    float* __restrict__ qbuf, float* __restrict__ kbuf, float* __restrict__ vbuf) {
  __shared__ float Alds[TM][TK + 1];
  __shared__ float Blds[TK][TNB];
  const int tn = blockIdx.x;            // 0..23
  const int m0 = blockIdx.y * TM;
  const int n0 = tn * TNB;
  const int tid = threadIdx.x;
  const int lane = tid & 31;
  const int wave = tid >> 5;
  const int wm = (wave & 3) * 32;       // 0,32,64,96
  const int wn = (wave >> 2) * 64;      // 0,64
  const int lrow = lane & 15;
  const int lhi = lane >> 4;            // 0 or 1

  v8f acc[2][4] = {};

  for (int k0 = 0; k0 < D_; k0 += TK) {
    __syncthreads();
    for (int g = tid; g < (TM * TK) / 4; g += 256) {   // 1024 float4 groups
      const int r = g >> 3;
      const int c = (g & 7) << 2;
      const float4 t =
          *reinterpret_cast<const float4*>(A + (size_t)(m0 + r) * D_ + k0 + c);
      Alds[r][c] = t.x; Alds[r][c + 1] = t.y; Alds[r][c + 2] = t.z; Alds[r][c + 3] = t.w;
    }
    for (int g = tid; g < (TK * TNB) / 4; g += 256) {  // 1024 float4 groups
      const int r = g >> 5;
      const int c = (g & 31) << 2;
      const float4 t = *reinterpret_cast<const float4*>(
          W + (size_t)(k0 + r) * (3 * D_) + n0 + c);
      *reinterpret_cast<float4*>(&Blds[r][c]) = t;
    }
    __syncthreads();
#pragma unroll
    for (int kk = 0; kk < TK; kk += 4) {
      const int kb = kk + lhi * 2;
      v2f a[2], b[4];
#pragma unroll
      for (int i = 0; i < 2; i++) {
        const int row = wm + i * 16 + lrow;
        a[i].x = Alds[row][kb];
        a[i].y = Alds[row][kb + 1];
      }
#pragma unroll
      for (int j = 0; j < 4; j++) {
        const int col = wn + j * 16 + lrow;
        b[j].x = Blds[kb][col];
        b[j].y = Blds[kb + 1][col];
      }
#pragma unroll
      for (int i = 0; i < 2; i++)
#pragma unroll
        for (int j = 0; j < 4; j++) acc[i][j] = wmma_f32(a[i], b[j], acc[i][j]);
    }
  }

  const int part = tn >> 3;  // 0=q,1=k,2=v (tile never crosses a part boundary)
  float* dst = (part == 0) ? qbuf : (part == 1) ? kbuf : vbuf;
  const bool do_phi = (part < 2);
#pragma unroll
  for (int i = 0; i < 2; i++)
#pragma unroll
    for (int j = 0; j < 4; j++)
#pragma unroll
      for (int r = 0; r < 8; r++) {
        const int rowl = wm + i * 16 + r + lhi * 8;
        const int colLocal = wn + j * 16 + lrow;     // 0..127
        const int h = (tn * 2 + (colLocal >> 6)) & 15;
        const int d = colLocal & 63;
        const int m = m0 + rowl;
        const int bb = m >> 12;
        const int l = m & (L_ - 1);
        float val = acc[i][j][r];
        if (do_phi) val = (val > 0.0f) ? (val + 1.0f) : __expf(val);
        dst[((size_t)((bb * H_ + h) * L_ + l)) * HD_ + d] = val;
      }
}

// ---------------------------------------------------------------------------
// 3) kv = phi(K)^T V per (b,h), accumulated over L chunks with f32 atomics.
//    Also accumulates ksum = sum_l phi(k)[l,:].
//    Block: one (b,h) x one 512-row L chunk. Output 64x64; 8 waves 4(M)x2(N).
// ---------------------------------------------------------------------------
__global__ __launch_bounds__(256) void kv_kernel(const float* __restrict__ kbuf,
                                                 const float* __restrict__ vbuf,
                                                 float* __restrict__ kv,
                                                 float* __restrict__ ksum) {
  __shared__ float Klds[TK][HD_ + 1];
  __shared__ float Vlds[TK][HD_ + 1];
  const int bh = blockIdx.x;
  const int l0 = blockIdx.y * KVCHUNK;
  const float* kp = kbuf + (size_t)bh * L_ * HD_;
  const float* vp = vbuf + (size_t)bh * L_ * HD_;
  const int tid = threadIdx.x;
  const int lane = tid & 31;
  const int wave = tid >> 5;
  const int wm = (wave & 3) * 16;   // 0..48
  const int wn = (wave >> 2) * 32;  // 0,32
  const int lrow = lane & 15;
  const int lhi = lane >> 4;

  v8f acc[2] = {};
  float ks = 0.0f;

  for (int kc = 0; kc < KVCHUNK; kc += TK) {
    __syncthreads();
    for (int g = tid; g < (TK * HD_) / 4; g += 256) {  // 512 groups
      const int r = g >> 4;
      const int c = (g & 15) << 2;
      const float4 tk =
          *reinterpret_cast<const float4*>(kp + (size_t)(l0 + kc + r) * HD_ + c);
      const float4 tv =
          *reinterpret_cast<const float4*>(vp + (size_t)(l0 + kc + r) * HD_ + c);
      Klds[r][c] = tk.x; Klds[r][c + 1] = tk.y; Klds[r][c + 2] = tk.z; Klds[r][c + 3] = tk.w;
      Vlds[r][c] = tv.x; Vlds[r][c + 1] = tv.y; Vlds[r][c + 2] = tv.z; Vlds[r][c + 3] = tv.w;
    }
    __syncthreads();
    if (tid < HD_) {
#pragma unroll
      for (int r = 0; r < TK; r++) ks += Klds[r][tid];
    }
#pragma unroll
    for (int kk = 0; kk < TK; kk += 4) {
      const int kb = kk + lhi * 2;
      v2f a;                               // A = k^T : A[d][l] = k[l][d]
      a.x = Klds[kb][wm + lrow];
      a.y = Klds[kb + 1][wm + lrow];
#pragma unroll
      for (int j = 0; j < 2; j++) {
        const int col = wn + j * 16 + lrow;
        v2f b;
        b.x = Vlds[kb][col];
        b.y = Vlds[kb + 1][col];
        acc[j] = wmma_f32(a, b, acc[j]);
      }
    }
  }
#pragma unroll
  for (int j = 0; j < 2; j++)
#pragma unroll
    for (int r = 0; r < 8; r++) {
      const int row = wm + r + lhi * 8;
      const int col = wn + j * 16 + lrow;
      atomicAdd(&kv[(size_t)bh * HD_ * HD_ + row * HD_ + col], acc[j][r]);
    }
  if (tid < HD_) atomicAdd(&ksum[bh * HD_ + tid], ks);
}

// ---------------------------------------------------------------------------
// 4) out = (q @ kv) / (q . ksum + eps), written back to [B,L,D] layout.
//    Block: one (b,h) x 128 L-rows. 8 waves, each one 16x64 strip (4 WMMA).
// ---------------------------------------------------------------------------
__global__ __launch_bounds__(256) void attn_kernel(const float* __restrict__ qbuf,
                                                   const float* __restrict__ kv,
                                                   const float* __restrict__ ksum,
                                                   float* __restrict__ attn) {
  __shared__ float KVlds[HD_][HD_ + 1];
  __shared__ float Qlds[TM][TK + 1];
  __shared__ float ks_lds[HD_];
  __shared__ float denom[TM];
  const int bh = blockIdx.x;
  const int l0 = blockIdx.y * TM;
  const int bb = bh >> 4;
  const int h = bh & 15;
  const float* qp = qbuf + (size_t)bh * L_ * HD_;
  const int tid = threadIdx.x;
  const int lane = tid & 31;
  const int wave = tid >> 5;
  const int mo = wave * 16;
  const int lrow = lane & 15;
  const int lhi = lane >> 4;

  for (int g = tid; g < (HD_ * HD_) / 4; g += 256) {  // 1024 groups
    const int r = g >> 4;
    const int c = (g & 15) << 2;
    const float4 t =
        *reinterpret_cast<const float4*>(kv + (size_t)bh * HD_ * HD_ + r * HD_ + c);
    KVlds[r][c] = t.x; KVlds[r][c + 1] = t.y; KVlds[r][c + 2] = t.z; KVlds[r][c + 3] = t.w;
  }
  if (tid < HD_) ks_lds[tid] = ksum[bh * HD_ + tid];

  v8f acc[4] = {};
  float dacc = 0.0f;

  for (int k0 = 0; k0 < HD_; k0 += TK) {
    __syncthreads();
    for (int g = tid; g < (TM * TK) / 4; g += 256) {
      const int r = g >> 3;
      const int c = (g & 7) << 2;
      const float4 t =
          *reinterpret_cast<const float4*>(qp + (size_t)(l0 + r) * HD_ + k0 + c);
      Qlds[r][c] = t.x; Qlds[r][c + 1] = t.y; Qlds[r][c + 2] = t.z; Qlds[r][c + 3] = t.w;
    }
    __syncthreads();
    if (tid < TM) {
#pragma unroll
      for (int j = 0; j < TK; j++) dacc += Qlds[tid][j] * ks_lds[k0 + j];
    }
#pragma unroll
    for (int kk = 0; kk < TK; kk += 4) {
      const int kb = kk + lhi * 2;
      v2f a;
      a.x = Qlds[mo + lrow][kb];
      a.y = Qlds[mo + lrow][kb + 1];
#pragma unroll
      for (int j = 0; j < 4; j++) {
        const int col = j * 16 + lrow;
        const int kr = k0 + kb;
        v2f b;
        b.x = KVlds[kr][col];
        b.y = KVlds[kr + 1][col];
        acc[j] = wmma_f32(a, b, acc[j]);
      }
    }
  }
  if (tid < TM) denom[tid] = dacc + 1e-6f;
  __syncthreads();
#pragma unroll
  for (int j = 0; j < 4; j++)
#pragma unroll
    for (int r = 0; r < 8; r++) {
      const int row = mo + r + lhi * 8;
      const int col = j * 16 + lrow;
      const float val = acc[j][r] / denom[row];
      attn[((size_t)(bb * L_ + l0 + row)) * D_ + h * HD_ + col] = val;
    }
}

// ---------------------------------------------------------------------------
// 5) out = residual + attn @ Wout  ([16384,1024] @ [1024,1024]), 128x128 tiles
// ---------------------------------------------------------------------------
__global__ __launch_bounds__(256) void out_gemm_kernel(const float* __restrict__ A,
                                                       const float* __restrict__ W,
                                                       const float* __restrict__ resid,
                                                       float* __restrict__ out) {
  __shared__ float Alds[TM][TK + 1];
  __shared__ float Blds[TK][TNB];
  const int m0 = blockIdx.y * TM;
  const int n0 = blockIdx.x * TNB;
  const int tid = threadIdx.x;
  const int lane = tid & 31;
  const int wave = tid >> 5;
  const int wm = (wave & 3) * 32;
  const int wn = (wave >> 2) * 64;
  const int lrow = lane & 15;
  const int lhi = lane >> 4;

  v8f acc[2][4] = {};

  for (int k0 = 0; k0 < D_; k0 += TK) {
    __syncthreads();
    for (int g = tid; g < (TM * TK) / 4; g += 256) {
      const int r = g >> 3;
      const int c = (g & 7) << 2;
      const float4 t =
          *reinterpret_cast<const float4*>(A + (size_t)(m0 + r) * D_ + k0 + c);
      Alds[r][c] = t.x; Alds[r][c + 1] = t.y; Alds[r][c + 2] = t.z; Alds[r][c + 3] = t.w;
    }
    for (int g = tid; g < (TK * TNB) / 4; g += 256) {
      const int r = g >> 5;
      const int c = (g & 31) << 2;
      const float4 t =
          *reinterpret_cast<const float4*>(W + (size_t)(k0 + r) * D_ + n0 + c);
      *reinterpret_cast<float4*>(&Blds[r][c]) = t;
    }
    __syncthreads();
#pragma unroll
    for (int kk = 0; kk < TK; kk += 4) {
      const int kb = kk + lhi * 2;
      v2f a[2], b[4];
#pragma unroll
      for (int i = 0; i < 2; i++) {
        const int row = wm + i * 16 + lrow;
        a[i].x = Alds[row][kb];
        a[i].y = Alds[row][kb + 1];
      }
#pragma unroll
      for (int j = 0; j < 4; j++) {
        const int col = wn + j * 16 + lrow;
        b[j].x = Blds[kb][col];
        b[j].y = Blds[kb + 1][col];
      }
#pragma unroll
      for (int i = 0; i < 2; i++)
#pragma unroll
        for (int j = 0; j < 4; j++) acc[i][j] = wmma_f32(a[i], b[j], acc[i][j]);
    }
  }
#pragma unroll
  for (int i = 0; i < 2; i++)
#pragma unroll
    for (int j = 0; j < 4; j++)
#pragma unroll
      for (int r = 0; r < 8; r++) {
        const int m = m0 + wm + i * 16 + r + lhi * 8;
        const int gcol = n0 + wn + j * 16 + lrow;
        out[(size_t)m * D_ + gcol] = acc[i][j][r] + resid[(size_t)m * D_ + gcol];
      }
}

// ---------------------------------------------------------------------------
extern "C" void kernel_launch(void* const* d_in, const int* in_sizes, int n_in,
                              void* d_out, int out_size, void* d_ws, size_t ws_size,
                              hipStream_t stream) {
  const float* x = (const float*)d_in[0];
  const float* gamma = (const float*)d_in[1];
  const float* beta = (const float*)d_in[2];
  const float* Wqkv = (const float*)d_in[3];
  const float* Wout = (const float*)d_in[4];
  float* out = (float*)d_out;
  float* ws = (float*)d_ws;

  const size_t NEL = (size_t)M_TOTAL * D_;  // 16,777,216 per tensor
  float* xn_attn = ws;                      // reused: xn then attn output
  float* qb = ws + NEL;
  float* kb = ws + 2 * NEL;
  float* vb = ws + 3 * NEL;
  float* kvb = ws + 4 * NEL;                       // B*H*64*64 = 262144
  float* ksb = kvb + (size_t)B_ * H_ * HD_ * HD_;  // B*H*64 = 4096

  const int nzero = B_ * H_ * HD_ * HD_ + B_ * H_ * HD_;
  zero_kernel<<<(nzero + 255) / 256, 256, 0, stream>>>(kvb, nzero);
  ln_kernel<<<M_TOTAL, 256, 0, stream>>>(x, gamma, beta, xn_attn);
  qkv_gemm_kernel<<<dim3((3 * D_) / TNB, M_TOTAL / TM), 256, 0, stream>>>(
      xn_attn, Wqkv, qb, kb, vb);
  kv_kernel<<<dim3(B_ * H_, L_ / KVCHUNK), 256, 0, stream>>>(kb, vb, kvb, ksb);
  attn_kernel<<<dim3(B_ * H_, L_ / TM), 256, 0, stream>>>(qb, kvb, ksb, xn_attn);
  out_gemm_kernel<<<dim3(D_ / TNB, M_TOTAL / TM), 256, 0, stream>>>(
      xn_attn, Wout, x, out);
}